// Anchor_Target_Layer_60404420051304
// MI455X (gfx1250) — compile-verified
//
#include <hip/hip_runtime.h>
#include <stdint.h>

#define POS_THR 0.5f
#define NEG_THR 0.3f
#define POS_BOUND 128   // SAMPLE_MAX_NUM * NP_RATE

typedef float v2f __attribute__((ext_vector_type(2)));
typedef float v8f __attribute__((ext_vector_type(8)));

__device__ __forceinline__ unsigned long long pack_iou(float iou, int idx) {
    // iou >= 0 so float bits are order-preserving; ~idx so ties pick smaller index
    return ((unsigned long long)__float_as_uint(iou) << 32) |
           (unsigned long long)(0xFFFFFFFFu - (unsigned)idx);
}

__device__ __forceinline__ float smooth_l1(float x) {
    float ax = fabsf(x);
    return ax < 1.0f ? 0.5f * x * x : ax - 0.5f;
}

// ---------------- kernel 1: zero the per-GT packed-argmax slots ----------------
__global__ void k_init(unsigned long long* gtPacked, int G) {
    int j = threadIdx.x;
    if (j < G) gtPacked[j] = 0ull;
}

// ---------------- kernel 2: streaming IoU pass (bandwidth-bound) ----------------
// meta[i] = flags | (argmax_j << 8); flags: bit0 = max>POS_THR, bit1 = max<NEG_THR
// (bit2 = forced-pos, set later by k_fixup). Also emits per-block predicate
// counts for the parallel ordered selection, and per-GT column max via LDS
// ds_max_u64 pre-reduction + one global u64 atomic max per GT per block.
__global__ void k_main(const float* __restrict__ anchors,
                       const float* __restrict__ gts,
                       uint16_t* __restrict__ meta,
                       unsigned long long* __restrict__ gtPacked,
                       int* __restrict__ posBlk, int* __restrict__ negBlk,
                       int A, int G) {
    __shared__ float gx0[64], gy0[64], gx1[64], gy1[64], gar[64];
    __shared__ unsigned long long col[64];
    __shared__ int cw0[8], cw1[8];
    int tid = threadIdx.x;
    int lane = tid & 31, wave = tid >> 5;
    if (tid < G) {
        float4 g = ((const float4*)gts)[tid];
        gx0[tid] = g.x; gy0[tid] = g.y; gx1[tid] = g.z; gy1[tid] = g.w;
        gar[tid] = (g.z - g.x) * (g.w - g.y);
        col[tid] = 0ull;
    }
    __syncthreads();

    int i = blockIdx.x * blockDim.x + tid;
    bool valid = i < A;
    unsigned f = 0u;
    if (valid) {
        // gfx1250: lowers to global_prefetch_b8 — pull the stream ahead
        int pf = i + 4096; if (pf >= A) pf = A - 1;
        __builtin_prefetch(&anchors[4 * pf], 0, 1);

        float4 a = ((const float4*)anchors)[i];
        float aarea = (a.z - a.x) * (a.w - a.y);
        float best = -1.0f; int bestj = 0;
        for (int j = 0; j < G; ++j) {
            float lx = fmaxf(a.x, gx0[j]);
            float ly = fmaxf(a.y, gy0[j]);
            float rx = fminf(a.z, gx1[j]);
            float ry = fminf(a.w, gy1[j]);
            float w = fmaxf(rx - lx, 0.0f);
            float h = fmaxf(ry - ly, 0.0f);
            float inter = w * h;
            float iou = inter / (aarea + gar[j] - inter);
            if (iou > best) { best = iou; bestj = j; }   // strict >: first max wins
            unsigned long long p = pack_iou(iou, i);
            if (p > *(volatile unsigned long long*)&col[j])
                atomicMax(&col[j], p);                    // ds_max_u64
        }
        if (best > POS_THR) f |= 1u;
        if (best < NEG_THR) f |= 2u;
        meta[i] = (uint16_t)(f | ((unsigned)bestj << 8));
    }

    // per-block predicate counts (pre-fixup): wave32 ballot + popc
    unsigned b0 = __builtin_amdgcn_ballot_w32((f & 1u) != 0u);
    unsigned b1 = __builtin_amdgcn_ballot_w32((f & 2u) != 0u);
    if (lane == 0) { cw0[wave] = __popc(b0); cw1[wave] = __popc(b1); }
    __syncthreads();
    if (tid == 0) {
        int s0 = 0, s1 = 0;
        for (int w = 0; w < 8; ++w) { s0 += cw0[w]; s1 += cw1[w]; }
        posBlk[blockIdx.x] = s0;
        negBlk[blockIdx.x] = s1;
    }
    if (tid < G) atomicMax(&gtPacked[tid], col[tid]);     // global_atomic_max_u64
}

// ---------------- kernel 3: decode gt_argmax, force pos bits, patch counts ----
// Duplicate gt_argmax anchors adjust counts exactly once (atomicOr dedup),
// matching the idempotent pos_mask.at[gt_argmax].set(True).
__global__ void k_fixup(const unsigned long long* __restrict__ gtPacked,
                        uint32_t* __restrict__ metaWords,
                        int* __restrict__ posBlk, int* __restrict__ negBlk,
                        int G) {
    int j = threadIdx.x;
    if (j >= G) return;
    unsigned idx = 0xFFFFFFFFu - (unsigned)(gtPacked[j] & 0xFFFFFFFFull);
    unsigned word  = idx >> 1;
    unsigned shift = (idx & 1u) * 16u;
    unsigned old = atomicOr(&metaWords[word], 4u << shift);
    unsigned mOld = (old >> shift) & 0xFFFFu;
    if ((mOld & 4u) == 0u) {                 // first GT to force this anchor
        int b = (int)(idx >> 8);             // block of 256 containing idx
        if ((mOld & 1u) == 0u) atomicAdd(&posBlk[b], 1);   // newly positive
        if ((mOld & 2u) != 0u) atomicAdd(&negBlk[b], -1);  // no longer negative
    }
}

// ---------------- kernel 4: exclusive scan over per-block counts ----------------
__global__ void k_prefix(const int* __restrict__ posBlk,
                         const int* __restrict__ negBlk,
                         int* __restrict__ posBase, int* __restrict__ negBase,
                         int* __restrict__ counts, int nb) {
    __shared__ int sp[1024], sn[1024];
    __shared__ int runP, runN;
    int tid = threadIdx.x;
    if (tid == 0) { runP = 0; runN = 0; }
    __syncthreads();
    for (int c0 = 0; c0 < nb; c0 += 1024) {
        int i = c0 + tid;
        int vp = (i < nb) ? posBlk[i] : 0;
        int vn = (i < nb) ? negBlk[i] : 0;
        sp[tid] = vp; sn[tid] = vn;
        __syncthreads();
        for (int off = 1; off < 1024; off <<= 1) {   // Hillis-Steele inclusive
            int ap = (tid >= off) ? sp[tid - off] : 0;
            int an = (tid >= off) ? sn[tid - off] : 0;
            __syncthreads();
            sp[tid] += ap; sn[tid] += an;
            __syncthreads();
        }
        if (i < nb) {
            posBase[i] = runP + sp[tid] - vp;        // exclusive
            negBase[i] = runN + sn[tid] - vn;
        }
        __syncthreads();
        if (tid == 0) { runP += sp[1023]; runN += sn[1023]; }
        __syncthreads();
    }
    if (tid == 0) { counts[0] = runP; counts[1] = runN; }
}

// ---------------- kernel 5: parallel ordered emit of first-128 indices ----------
__global__ void k_emit(const uint16_t* __restrict__ meta,
                       const int* __restrict__ posBase,
                       const int* __restrict__ negBase,
                       int* __restrict__ posIdx, int* __restrict__ negIdx,
                       int A) {
    __shared__ int wp[8], wn[8], wpB[8], wnB[8];
    int b = blockIdx.x, tid = threadIdx.x;
    int pb = posBase[b], nb0 = negBase[b];
    if (pb >= POS_BOUND && nb0 >= POS_BOUND) return;  // all later: nothing to emit
    int lane = tid & 31, wave = tid >> 5;
    int i = b * 256 + tid;
    bool valid = i < A;
    unsigned m = valid ? (unsigned)meta[i] : 0u;
    bool pos = valid && (((m & 1u) != 0u) || ((m & 4u) != 0u));
    bool neg = valid && (((m & 2u) != 0u) && ((m & 4u) == 0u));
    unsigned bp = __builtin_amdgcn_ballot_w32(pos);
    unsigned bn = __builtin_amdgcn_ballot_w32(neg);
    if (lane == 0) { wp[wave] = __popc(bp); wn[wave] = __popc(bn); }
    __syncthreads();
    if (tid == 0) {
        int rp = 0, rn = 0;
        for (int w = 0; w < 8; ++w) {
            wpB[w] = rp; rp += wp[w];
            wnB[w] = rn; rn += wn[w];
        }
    }
    __syncthreads();
    unsigned lmask = (1u << lane) - 1u;
    if (pos) {
        int p = pb + wpB[wave] + __popc(bp & lmask);
        if (p < POS_BOUND) posIdx[p] = i;
    }
    if (neg) {
        int p = nb0 + wnB[wave] + __popc(bn & lmask);
        if (p < POS_BOUND) negIdx[p] = i;
    }
}

// ---------------- kernel 6: losses + WMMA sum-reduction ----------------
__global__ void k_loss(const float* __restrict__ score,
                       const float* __restrict__ reg,
                       const float* __restrict__ anchors,
                       const float* __restrict__ gts,
                       const uint16_t* __restrict__ meta,
                       const int* __restrict__ posIdx,
                       const int* __restrict__ negIdx,
                       const int* __restrict__ counts,
                       float* __restrict__ out) {
    __shared__ float buf[256];
    int tid = threadIdx.x;
    int posTotal = counts[0];
    int negTotal = counts[1];
    int B = posTotal < POS_BOUND ? posTotal : POS_BOUND;  // runtime_pos (both branches)
    bool bug = posTotal >= POS_BOUND;
    int negStored = negTotal < POS_BOUND ? negTotal : POS_BOUND;
    int nn = negStored < B ? negStored : B;               // neg_bound == runtime_pos
    float invNp = 1.0f / (float)B;
    float invNn = 1.0f / (float)nn;

    float partial = 0.0f;
    if (tid < B) {
        int idx = posIdx[tid];
        float4 a = ((const float4*)anchors)[idx];
        float ce;
        if (bug) { // reference bug branch: anchors used as 4-way logits, target 0
            float mx = fmaxf(fmaxf(a.x, a.y), fmaxf(a.z, a.w));
            float s = expf(a.x - mx) + expf(a.y - mx) + expf(a.z - mx) + expf(a.w - mx);
            ce = (mx + logf(s)) - a.x;
        } else {
            float s0 = score[2 * idx], s1 = score[2 * idx + 1];
            float mx = fmaxf(s0, s1);
            float s = expf(s0 - mx) + expf(s1 - mx);
            ce = (mx + logf(s)) - s0;
        }
        partial += ce * invNp;

        int mj = (int)(meta[idx] >> 8);                   // anchor_argmax (mapper)
        float4 g = ((const float4*)gts)[mj];
        float aw = a.z - a.x, ah = a.w - a.y;
        float acx = a.x + aw * 0.5f, acy = a.y + ah * 0.5f;
        float gw = g.z - g.x, gh = g.w - g.y;
        float gcx = g.x + gw * 0.5f, gcy = g.y + gh * 0.5f;
        float t0 = (gcx - acx) / aw;
        float t1 = (gcy - acy) / ah;
        float t2 = logf(gw / aw);
        float t3 = logf(gh / ah);
        float4 r = ((const float4*)reg)[idx];
        float sl = smooth_l1(r.x - t0) + smooth_l1(r.y - t1) +
                   smooth_l1(r.z - t2) + smooth_l1(r.w - t3);
        partial += sl * invNp;                            // / runtime_pos
    }
    if (tid < nn) {
        int idx = negIdx[tid];
        float s0 = score[2 * idx], s1 = score[2 * idx + 1];
        float mx = fmaxf(s0, s1);
        float s = expf(s0 - mx) + expf(s1 - mx);
        float ce = (mx + logf(s)) - s1;                   // target 1
        partial += ce * invNn;
    }
    buf[tid] = partial;
    __syncthreads();

    // Deterministic 256 -> 1 sum via V_WMMA_F32_16X16X4_F32 with B = ones:
    // D[m][n] = sum_k A[m][k]; total = sum over a D column = sum of all A values.
    if (tid < 32) {                                       // wave 0 only: EXEC all ones
        int lane = tid;
        int mrow = lane & 15;
        int kb = (lane < 16) ? 0 : 2;                     // A 16x4 f32 VGPR layout
        v8f c = {};
        v2f ones; ones.x = 1.0f; ones.y = 1.0f;
        for (int ch = 0; ch < 4; ++ch) {
            v2f av;
            av.x = buf[ch * 64 + mrow * 4 + kb];
            av.y = buf[ch * 64 + mrow * 4 + kb + 1];
            c = __builtin_amdgcn_wmma_f32_16x16x4_f32(
                    false, av, false, ones, (short)0, c, false, false);
        }
        float s = c[0] + c[1] + c[2] + c[3] + c[4] + c[5] + c[6] + c[7];
        float o = __shfl(s, 16, 32);                      // rows 8..15 live in lane 16
        if (lane == 0) out[0] = s + o;
    }
}

extern "C" void kernel_launch(void* const* d_in, const int* in_sizes, int n_in,
                              void* d_out, int out_size, void* d_ws, size_t ws_size,
                              hipStream_t stream) {
    const float* score   = (const float*)d_in[0];   // (A,2)
    const float* reg     = (const float*)d_in[1];   // (A,4)
    const float* anchors = (const float*)d_in[2];   // (A,4)
    const float* gts     = (const float*)d_in[3];   // (G,4)
    int A = in_sizes[2] / 4;
    int G = in_sizes[3] / 4;
    int nb = (A + 255) / 256;

    char* ws = (char*)d_ws;
    unsigned long long* gtPacked = (unsigned long long*)ws;         // 64*8 = 512 B
    int* counts = (int*)(ws + 512);                                 // 2 ints
    int* posIdx = (int*)(ws + 1024);                                // 512 B
    int* negIdx = (int*)(ws + 1536);                                // 512 B
    size_t off = 2048;
    int* posBlk  = (int*)(ws + off); off += (size_t)4 * nb;
    int* negBlk  = (int*)(ws + off); off += (size_t)4 * nb;
    int* posBase = (int*)(ws + off); off += (size_t)4 * nb;
    int* negBase = (int*)(ws + off); off += (size_t)4 * nb;
    off = (off + 255) & ~(size_t)255;
    uint16_t* meta = (uint16_t*)(ws + off);                         // 2*A bytes
    float* out = (float*)d_out;

    k_init  <<<1, 64, 0, stream>>>(gtPacked, G);
    k_main  <<<nb, 256, 0, stream>>>(anchors, gts, meta, gtPacked,
                                     posBlk, negBlk, A, G);
    k_fixup <<<1, 64, 0, stream>>>(gtPacked, (uint32_t*)meta, posBlk, negBlk, G);
    k_prefix<<<1, 1024, 0, stream>>>(posBlk, negBlk, posBase, negBase, counts, nb);
    k_emit  <<<nb, 256, 0, stream>>>(meta, posBase, negBase, posIdx, negIdx, A);
    k_loss  <<<1, 256, 0, stream>>>(score, reg, anchors, gts, meta,
                                    posIdx, negIdx, counts, out);
}